// CombinedGaussianLoss_53291954208916
// MI455X (gfx1250) — compile-verified
//
#include <hip/hip_runtime.h>
#include <math.h>

#define PIMG 16
#define HH 512
#define WW 512
#define NTOT (PIMG * HH * WW)
#define THRESH 0.1f
#define MAX_ITERS 128

typedef __attribute__((ext_vector_type(2))) float v2f;
typedef __attribute__((ext_vector_type(8))) float v8f;

struct K9  { float k[9];  };
struct K40 { float k[40]; };   // zero-padded band table: k[15+t] = gauss[t], t=0..8

__device__ __forceinline__ int refl(int i, int n) {
  // numpy 'symmetric' (reflect including edge); radius 4 << n so one fold suffices
  if (i < 0) return -1 - i;
  if (i >= n) return 2 * n - 1 - i;
  return i;
}

// ---------------- Pass 1: Gaussian blur along H (axis 1), 4 px/thread ----------------
__global__ void blur_h(const float* __restrict__ gt, float* __restrict__ out, K9 kk) {
  int idx = blockIdx.x * blockDim.x + threadIdx.x;   // one thread per 4 consecutive pixels
  int g = idx << 2;
  int x = g & (WW - 1);
  int y = (g >> 9) & (HH - 1);
  int p = g >> 18;
  const float* img = gt + (size_t)p * (HH * WW);
  float a0 = 0.f, a1 = 0.f, a2 = 0.f, a3 = 0.f;
#pragma unroll
  for (int t = 0; t < 9; ++t) {                      // t constant after unroll -> sgpr kernarg reads
    int yy = refl(y + t - 4, HH);
    const float4 v = *(const float4*)(img + yy * WW + x);
    float w = kk.k[t];
    a0 += w * v.x; a1 += w * v.y; a2 += w * v.z; a3 += w * v.w;
  }
  float4 r; r.x = a0; r.y = a1; r.z = a2; r.w = a3;
  *(float4*)(out + g) = r;
}

// ------- Pass 2: blur along W as banded-matrix WMMA GEMM + threshold -------
// One wave computes a 16x16 output tile: D(16x16) = A(16x24 input slab) x B(24x16 band),
// chunked as six v_wmma_f32_16x16x4_f32 accumulations. Branch-free B via LDS band table.
__global__ void blur_w_wmma(const float* __restrict__ in, unsigned char* __restrict__ mask, K40 kb) {
  __shared__ float kpad[40];
  __shared__ float tile[8][16 * 25];  // row stride 25 -> conflict-free LDS reads
  if (threadIdx.x < 40) kpad[threadIdx.x] = kb.k[threadIdx.x];

  int w = threadIdx.x >> 5;
  int lane = threadIdx.x & 31;
  int T = blockIdx.x * 8 + w;          // tile id: 16 imgs * 32 * 32 tiles
  int p = T >> 10;
  int rem = T & 1023;
  int y0 = (rem >> 5) << 4;
  int c0 = (rem & 31) << 4;
  const float* img = in + (size_t)p * (HH * WW);

  // fill the 16x24 input slab (columns c0-4 .. c0+19, symmetric edges)
  for (int i = lane; i < 16 * 24; i += 32) {
    int r = i / 24, s = i % 24;
    tile[w][r * 25 + s] = img[(y0 + r) * WW + refl(c0 - 4 + s, WW)];
  }
  __syncthreads();

  int m = lane & 15;   // A row (image row offset) == B column (output col offset)
  int h = lane >> 4;   // lane half selects K pair {0,1} vs {2,3}
  v8f c = {0.f, 0.f, 0.f, 0.f, 0.f, 0.f, 0.f, 0.f};
#pragma unroll
  for (int q = 0; q < 6; ++q) {
    int s0 = 4 * q + 2 * h;
    v2f a, b;
    a.x = tile[w][m * 25 + s0];
    a.y = tile[w][m * 25 + s0 + 1];
    int d0 = s0 - m;                   // tap index = s - n; kpad is zero outside [15,23]
    b.x = kpad[d0 + 15];
    b.y = kpad[d0 + 16];
    c = __builtin_amdgcn_wmma_f32_16x16x4_f32(false, a, false, b, (short)0, c, false, false);
  }

  unsigned char* mimg = mask + (size_t)p * (HH * WW);
#pragma unroll
  for (int r = 0; r < 8; ++r) {        // D layout: VGPR r -> M = r + 8*h, N = lane&15
    int y = y0 + r + 8 * h;
    mimg[y * WW + c0 + m] = (c[r] > THRESH) ? (unsigned char)1 : (unsigned char)0;
  }
}

// ---------------- Pass 3: Zhang-Suen thinning sub-step, 4 px/thread ----------------
__global__ void zs_substep(const unsigned char* __restrict__ src, unsigned char* __restrict__ dst,
                           int step, int iter, int* __restrict__ changed) {
  // converged => both ping-pong buffers already hold the fixed point; skip.
  if (iter > 0 && changed[iter - 1] == 0) return;
  int idx = blockIdx.x * blockDim.x + threadIdx.x;   // 4 consecutive pixels per thread
  int g = idx << 2;
  int x = g & (WW - 1);                // multiple of 4 -> dword aligned
  int y = (g >> 9) & (HH - 1);
  int p = g >> 18;
  const unsigned char* img = src + (size_t)p * (HH * WW);

  // 3 rows x {left,center,right} aligned dwords; zero outside image (reference zero-pads)
  unsigned rw[3][3];
#pragma unroll
  for (int r = 0; r < 3; ++r) {
    int yy = y + r - 1;
    bool yok = (yy >= 0) && (yy < HH);
    const unsigned* rowp = (const unsigned*)(img + yy * WW);
    rw[r][0] = (yok && x >= 4)     ? rowp[(x >> 2) - 1] : 0u;
    rw[r][1] = yok                 ? rowp[(x >> 2)]     : 0u;
    rw[r][2] = (yok && x + 4 < WW) ? rowp[(x >> 2) + 1] : 0u;
  }
  auto bget = [&](int r, int k) -> int {
    if (k < 0)  return (int)(rw[r][0] >> 24) & 0xFF;
    if (k < 4)  return (int)(rw[r][1] >> (8 * k)) & 0xFF;
    return (int)(rw[r][2]) & 0xFF;
  };

  unsigned outw = 0;
  int anyDel = 0;
#pragma unroll
  for (int j = 0; j < 4; ++j) {
    int cen = bget(1, j);
    int del = 0;
    if (cen) {
      int p9 = bget(0, j - 1), p2 = bget(0, j), p3 = bget(0, j + 1);
      int p8 = bget(1, j - 1),                  p4 = bget(1, j + 1);
      int p7 = bget(2, j - 1), p6 = bget(2, j), p5 = bget(2, j + 1);
      int Bn = p2 + p3 + p4 + p5 + p6 + p7 + p8 + p9;
      int A = (!p2 && p3) + (!p3 && p4) + (!p4 && p5) + (!p5 && p6) +
              (!p6 && p7) + (!p7 && p8) + (!p8 && p9) + (!p9 && p2);
      int c1, c2;
      if (step == 0) { c1 = !(p2 && p4 && p6); c2 = !(p4 && p6 && p8); }
      else           { c1 = !(p2 && p4 && p8); c2 = !(p2 && p6 && p8); }
      del = (Bn >= 2) && (Bn <= 6) && (A == 1) && c1 && c2;
    }
    anyDel |= del;
    outw |= (unsigned)(del ? 0 : cen) << (8 * j);
  }
  *(unsigned*)(dst + g) = outw;
  unsigned long long bal = __ballot(anyDel);   // wave32: at most one atomic per wave
  if (bal != 0ull && ((threadIdx.x & 31) == 0)) atomicAdd(changed + iter, 1);
}

__global__ void init_changed(int* __restrict__ changed) {
  if (threadIdx.x < MAX_ITERS) changed[threadIdx.x] = 0;
}

// ---------------- Pass 4: deterministic two-stage loss reduction (b128 loads) ----------------
__global__ void loss_partial(const float* __restrict__ pred, const float* __restrict__ gt,
                             const unsigned char* __restrict__ mask, float* __restrict__ partials) {
  __shared__ float s0s[256], s1s[256];
  int tid = threadIdx.x;
  float s0 = 0.f, s1 = 0.f;
  const int N4 = NTOT / 4;
  for (int i = blockIdx.x * blockDim.x + tid; i < N4; i += gridDim.x * blockDim.x) {
    float4 pv = ((const float4*)pred)[i];
    float4 gv = ((const float4*)gt)[i];
    unsigned mw = ((const unsigned*)mask)[i];
    float d, d2;
    d = pv.x - gv.x; d2 = d * d; s0 += d2; if (mw & 0xFFu)         s1 += d2;
    d = pv.y - gv.y; d2 = d * d; s0 += d2; if ((mw >> 8) & 0xFFu)  s1 += d2;
    d = pv.z - gv.z; d2 = d * d; s0 += d2; if ((mw >> 16) & 0xFFu) s1 += d2;
    d = pv.w - gv.w; d2 = d * d; s0 += d2; if (mw >> 24)           s1 += d2;
  }
  s0s[tid] = s0; s1s[tid] = s1;
  __syncthreads();
  for (int s = 128; s > 0; s >>= 1) {
    if (tid < s) { s0s[tid] += s0s[tid + s]; s1s[tid] += s1s[tid + s]; }
    __syncthreads();
  }
  if (tid == 0) { partials[2 * blockIdx.x] = s0s[0]; partials[2 * blockIdx.x + 1] = s1s[0]; }
}

__global__ void loss_final(const float* __restrict__ partials, float* __restrict__ out, int nblocks) {
  __shared__ float s0s[256], s1s[256];
  int tid = threadIdx.x;
  float s0 = 0.f, s1 = 0.f;
  for (int i = tid; i < nblocks; i += 256) { s0 += partials[2 * i]; s1 += partials[2 * i + 1]; }
  s0s[tid] = s0; s1s[tid] = s1;
  __syncthreads();
  for (int s = 128; s > 0; s >>= 1) {
    if (tid < s) { s0s[tid] += s0s[tid + s]; s1s[tid] += s1s[tid + s]; }
    __syncthreads();
  }
  if (tid == 0) {
    float l2 = s0s[0] / (float)NTOT;
    float cl = s1s[0] / (float)NTOT;
    out[0] = 0.5f * l2 + 0.5f * cl;
  }
}

extern "C" void kernel_launch(void* const* d_in, const int* in_sizes, int n_in,
                              void* d_out, int out_size, void* d_ws, size_t ws_size,
                              hipStream_t stream) {
  const float* pred = (const float*)d_in[0];
  const float* gt   = (const float*)d_in[1];

  char* ws = (char*)d_ws;
  float*         blurtmp  = (float*)ws;                                  // 16 MB
  unsigned char* mA       = (unsigned char*)(ws + (size_t)NTOT * 4);     //  4 MB
  unsigned char* mB       = mA + NTOT;                                   //  4 MB
  int*           changed  = (int*)(ws + (size_t)NTOT * 4 + 2 * (size_t)NTOT);
  float*         partials = (float*)((char*)changed + 4096);             // 2048 floats

  // Gaussian taps in double, matching numpy (sigma=1, radius=4)
  K9 kk; K40 kb;
  {
    double kd[9], s = 0.0;
    for (int t = 0; t < 9; ++t) { double xx = (double)(t - 4); kd[t] = exp(-0.5 * xx * xx); s += kd[t]; }
    for (int t = 0; t < 40; ++t) kb.k[t] = 0.f;
    for (int t = 0; t < 9; ++t) { kk.k[t] = (float)(kd[t] / s); kb.k[15 + t] = kk.k[t]; }
  }

  init_changed<<<1, 128, 0, stream>>>(changed);
  blur_h<<<NTOT / 4 / 256, 256, 0, stream>>>(gt, blurtmp, kk);
  blur_w_wmma<<<(PIMG * 32 * 32) / 8, 256, 0, stream>>>(blurtmp, mA, kb);
  for (int it = 0; it < MAX_ITERS; ++it) {
    zs_substep<<<NTOT / 4 / 256, 256, 0, stream>>>(mA, mB, 0, it, changed);
    zs_substep<<<NTOT / 4 / 256, 256, 0, stream>>>(mB, mA, 1, it, changed);
  }
  loss_partial<<<1024, 256, 0, stream>>>(pred, gt, mA, partials);
  loss_final<<<1, 256, 0, stream>>>(partials, (float*)d_out, 1024);
}